// LowPassMAELoss_88321707475679
// MI455X (gfx1250) — compile-verified
//
#include <hip/hip_runtime.h>
#include <math.h>

// ---------------------------------------------------------------------------
// LowPassMAELoss on gfx1250 (MI455X):
//   loss = mean | butter6_lowpass(out) - butter6_lowpass(tgt) |
//        = mean | butter6_lowpass(out - tgt) |          (filter is linear)
//
// IIR as affine recurrence  s' = M s + q x,  y = s[0] + b0 x   (6-state DF2T).
// Parallelized by chunked affine scan + 16-step blocking mapped onto
// V_WMMA_F32_16X16X4_F32 (chained over K=24 = 16 inputs + 6 state + 2 pad).
// Memory bound: ~201 MB traffic -> ~8.7 us at 23.3 TB/s.
// ---------------------------------------------------------------------------

typedef __attribute__((ext_vector_type(2))) float v2f;
typedef __attribute__((ext_vector_type(8))) float v8f;

constexpr int ROWS         = 96;
constexpr int TLEN         = 131072;
constexpr int LCHUNK       = 512;          // samples per chunk (per WMMA column)
constexpr int WAVES_PER_ROW= 16;           // 16 waves x 16 columns x 512 = 131072
constexpr int NCHUNK_ROW   = 256;
constexpr int NCHUNK_TOT   = ROWS * NCHUNK_ROW;     // 24576
constexpr int TILES        = LCHUNK / 16;  // 32 16-step tiles per chunk
constexpr int NWAVES       = ROWS * WAVES_PER_ROW;  // 1536

// workspace layout (float offsets)
constexpr int TU_OFF = 0;                       // [16][24]  A-operand: [T | U | 0]
constexpr int VW_OFF = 384;                     // [16][24]  A-operand: [V | W | 0] (rows 6..15 zero)
constexpr int G_OFF  = 768;                     // [6][6]    M^512
constexpr int G8_OFF = 804;                     // [5][6][6] M^(4096*2^k), k=0..4
constexpr int H_OFF  = 1024;                    // [24576][6] zero-entry chunk exit states
constexpr int E_OFF  = H_OFF + NCHUNK_TOT * 6;  // [24576][6] chunk entry states
constexpr int P_OFF  = E_OFF + NCHUNK_TOT * 6;  // [1536]     per-wave |y| partial sums
// total = 297472 floats = ~1.19 MB of d_ws

// ---------------------------------------------------------------------------
// double 6x6 helpers (setup kernel only, 1 thread)
// ---------------------------------------------------------------------------
__device__ __forceinline__ void mm6(const double* A, const double* B, double* C) {
  for (int i = 0; i < 6; ++i)
    for (int j = 0; j < 6; ++j) {
      double acc = 0.0;
      for (int k = 0; k < 6; ++k) acc += A[i * 6 + k] * B[k * 6 + j];
      C[i * 6 + j] = acc;
    }
}
__device__ __forceinline__ void mv6(const double* A, const double* x, double* y) {
  for (int i = 0; i < 6; ++i) {
    double acc = 0.0;
    for (int k = 0; k < 6; ++k) acc += A[i * 6 + k] * x[k];
    y[i] = acc;
  }
}

// ---------------------------------------------------------------------------
// setup: Butterworth design + blocked-IIR constant tables, all in double.
// ---------------------------------------------------------------------------
__global__ void lp_setup(float* __restrict__ ws) {
  if (threadIdx.x != 0 || blockIdx.x != 0) return;
  const double PI = 3.14159265358979323846;
  // order 6, wn = 4000/24000 = 1/6; warped = 2*fs*tan(pi*wn/fs), fs=2
  double warped = 4.0 * tan(PI / 12.0);

  // three conjugate pole pairs -> bilinear -> real quadratics of a(z)
  double aq[3][3];
  double denprod = 1.0;  // prod |4 - p|^2  (real since conjugate pairs)
  for (int p = 0; p < 3; ++p) {
    double th  = PI * (double)(2 * p + 1) / 12.0;
    double pre = -warped * cos(th), pim = -warped * sin(th);
    double nr = 4.0 + pre, ni = pim, dr = 4.0 - pre, di = -pim;
    double dd = dr * dr + di * di;
    double zr = (nr * dr + ni * di) / dd;
    double zi = (ni * dr - nr * di) / dd;
    aq[p][0] = 1.0; aq[p][1] = -2.0 * zr; aq[p][2] = zr * zr + zi * zi;
    denprod *= dd;
  }
  double a[7] = {1, 0, 0, 0, 0, 0, 0}, tmp[7];
  int len = 1;
  for (int p = 0; p < 3; ++p) {
    for (int i = 0; i < 7; ++i) tmp[i] = 0.0;
    for (int i = 0; i < len; ++i)
      for (int j = 0; j < 3; ++j) tmp[i + j] += a[i] * aq[p][j];
    len += 2;
    for (int i = 0; i < len; ++i) a[i] = tmp[i];
  }
  double w2 = warped * warped;
  double kz = (w2 * w2 * w2) / denprod;              // k * Re(1/prod(4-p))
  const double binom[7] = {1, 6, 15, 20, 15, 6, 1};  // poly(-ones(6))
  double bb[7];
  for (int i = 0; i < 7; ++i) bb[i] = kz * binom[i];
  double b0 = bb[0];

  // DF2T state space: s' = M s + q x, y = s[0] + b0 x
  double M[36] = {0}, q[6];
  for (int i = 0; i < 6; ++i) {
    M[i * 6 + 0] = -a[i + 1];
    if (i < 5) M[i * 6 + i + 1] += 1.0;
    q[i] = bb[i + 1] - a[i + 1] * b0;
  }

  // first 16 impulse-response taps: h[0]=b0, h[d]=(M^(d-1) q)[0]
  double h[16], s[6], ns[6];
  h[0] = b0;
  for (int i = 0; i < 6; ++i) s[i] = q[i];
  for (int t = 1; t < 16; ++t) {
    h[t] = s[0];
    mv6(M, s, ns);
    for (int i = 0; i < 6; ++i) s[i] = ns[i];
  }
  // U[i][:] = row 0 of M^i ; after loop P = M^16 = W
  double P[36] = {0}, Pn[36], U[16][6];
  for (int i = 0; i < 6; ++i) P[i * 6 + i] = 1.0;
  for (int i = 0; i < 16; ++i) {
    for (int k = 0; k < 6; ++k) U[i][k] = P[k];
    mm6(M, P, Pn);
    for (int k = 0; k < 36; ++k) P[k] = Pn[k];
  }
  double Wm[36];
  for (int k = 0; k < 36; ++k) Wm[k] = P[k];
  // V[:,j] = M^(15-j) q
  double V[6][16], v[6];
  for (int k = 0; k < 6; ++k) v[k] = q[k];
  for (int j = 15; j >= 0; --j) {
    for (int k = 0; k < 6; ++k) V[k][j] = v[k];
    mv6(M, v, ns);
    for (int k = 0; k < 6; ++k) v[k] = ns[k];
  }

  // A-operand tables, row-major [16][24]
  for (int i = 0; i < 16; ++i)
    for (int k = 0; k < 24; ++k) {
      double tu = 0.0, vw = 0.0;
      if (k < 16) { int d = i - k; if (d >= 0) tu = h[d]; }
      else if (k < 22) tu = U[i][k - 16];
      if (i < 6) {
        if (k < 16) vw = V[i][k];
        else if (k < 22) vw = Wm[i * 6 + (k - 16)];
      }
      ws[TU_OFF + i * 24 + k] = (float)tu;
      ws[VW_OFF + i * 24 + k] = (float)vw;
    }

  // G = M^512 = (M^16)^32 : 5 squarings
  double G[36];
  for (int k = 0; k < 36; ++k) G[k] = Wm[k];
  for (int i = 0; i < 5; ++i) { mm6(G, G, Pn); for (int k = 0; k < 36; ++k) G[k] = Pn[k]; }
  for (int k = 0; k < 36; ++k) ws[G_OFF + k] = (float)G[k];
  // G^(8*2^i) = M^(4096*2^i), i=0..4 (scan multipliers)
  double Q[36];
  for (int k = 0; k < 36; ++k) Q[k] = G[k];
  for (int i = 0; i < 3; ++i) { mm6(Q, Q, Pn); for (int k = 0; k < 36; ++k) Q[k] = Pn[k]; }
  for (int k = 0; k < 36; ++k) ws[G8_OFF + k] = (float)Q[k];
  for (int i = 1; i < 5; ++i) {
    mm6(Q, Q, Pn);
    for (int k = 0; k < 36; ++k) { Q[k] = Pn[k]; ws[G8_OFF + i * 36 + k] = (float)Q[k]; }
  }
}

// ---------------------------------------------------------------------------
// WMMA tile helper notes (f32, wave32):
//  A (16x4): lane m=lane&15; vgpr v holds K = 4*wm + 2*(lane>>4) + v
//  B (4x16): lane n=lane&15; vgpr v holds K = 4*wm + 2*(lane>>4) + v (mirrors C/D half-split)
//  C/D (16x16): vgpr r: lanes 0-15 -> row r, lanes 16-31 -> row r+8, col = lane&15
// Chaining 6 WMMAs over K implements the 16x24 x 24x16 GEMM.
// ---------------------------------------------------------------------------

// k1: per-chunk zero-entry exit states (state GEMM only)
__global__ __launch_bounds__(256) void lp_chunkstate(const float* __restrict__ xo,
                                                     const float* __restrict__ xt,
                                                     float* __restrict__ ws) {
  const int wid  = (blockIdx.x * blockDim.x + threadIdx.x) >> 5;
  const int lane = threadIdx.x & 31;
  const int col  = lane & 15;
  const int hi   = lane >> 4;
  const int row  = wid / WAVES_PER_ROW;
  const int w    = wid % WAVES_PER_ROW;

  v2f AV[6];
  {
    const float* vw = ws + VW_OFF;
    #pragma unroll
    for (int wm = 0; wm < 6; ++wm) {
      int k0 = 4 * wm + 2 * hi;
      AV[wm].x = vw[col * 24 + k0];
      AV[wm].y = vw[col * 24 + k0 + 1];
    }
  }
  const long base = (long)row * TLEN + (long)(w * 16 + col) * LCHUNK;
  const float* po = xo + base;
  const float* pt = xt + base;

  float s0 = 0.f, s1 = 0.f, s2 = 0.f, s3 = 0.f, s4 = 0.f, s5 = 0.f;
  for (int t = 0; t < TILES; ++t) {
    const int tb = t * 16;
    v2f B[6];
    #pragma unroll
    for (int wm = 0; wm < 4; ++wm) {
      int k0 = tb + 4 * wm + 2 * hi;
      B[wm].x = po[k0]     - pt[k0];
      B[wm].y = po[k0 + 1] - pt[k0 + 1];
    }
    // inject per-column state as B rows 16..21 (22,23 zero padded)
    float t0 = __shfl(s0, col, 32), t1 = __shfl(s1, col, 32), t2 = __shfl(s2, col, 32);
    float t3 = __shfl(s3, col, 32), t4 = __shfl(s4, col, 32), t5 = __shfl(s5, col, 32);
    B[4].x = hi ? t2 : t0;   B[4].y = hi ? t3 : t1;
    B[5].x = hi ? 0.f : t4;  B[5].y = hi ? 0.f : t5;

    v8f d = {0.f, 0.f, 0.f, 0.f, 0.f, 0.f, 0.f, 0.f};
    #pragma unroll
    for (int wm = 0; wm < 6; ++wm)
      d = __builtin_amdgcn_wmma_f32_16x16x4_f32(false, AV[wm], false, B[wm],
                                                (short)0, d, false, false);
    s0 = d[0]; s1 = d[1]; s2 = d[2]; s3 = d[3]; s4 = d[4]; s5 = d[5];
  }
  if (lane < 16) {
    long c = (long)row * NCHUNK_ROW + w * 16 + col;
    float* hp = ws + H_OFF + c * 6;
    hp[0] = s0; hp[1] = s1; hp[2] = s2; hp[3] = s3; hp[4] = s4; hp[5] = s5;
  }
}

// k2: per-row affine Hillis-Steele scan over 256 chunk states -> entry states
__global__ __launch_bounds__(32) void lp_scan(float* __restrict__ ws) {
  const int row = blockIdx.x, lane = threadIdx.x;  // 32 lanes, 8 chunks each
  float G[36];
  for (int i = 0; i < 36; ++i) G[i] = ws[G_OFF + i];
  const int c0 = row * NCHUNK_ROW + lane * 8;

  // local block aggregate H = compose over own 8 chunks (zero entry)
  float b[6] = {0, 0, 0, 0, 0, 0};
  for (int j = 0; j < 8; ++j) {
    const float* hp = ws + H_OFF + (long)(c0 + j) * 6;
    float nb[6];
    for (int i = 0; i < 6; ++i) {
      float acc = hp[i];
      for (int k = 0; k < 6; ++k) acc += G[i * 6 + k] * b[k];
      nb[i] = acc;
    }
    for (int i = 0; i < 6; ++i) b[i] = nb[i];
  }
  // inclusive scan across lanes with multipliers G^(8*2^st)
  for (int st = 0; st < 5; ++st) {
    const int stride = 1 << st;
    float Pm[36];
    for (int i = 0; i < 36; ++i) Pm[i] = ws[G8_OFF + st * 36 + i];
    float nb[6];
    for (int i = 0; i < 6; ++i) nb[i] = __shfl_up(b[i], stride, 32);
    if (lane >= stride) {
      float t[6];
      for (int i = 0; i < 6; ++i) {
        float acc = b[i];
        for (int k = 0; k < 6; ++k) acc += Pm[i * 6 + k] * nb[k];
        t[i] = acc;
      }
      for (int i = 0; i < 6; ++i) b[i] = t[i];
    }
  }
  // exclusive = block entry state
  float e[6];
  for (int i = 0; i < 6; ++i) {
    float pv = __shfl_up(b[i], 1, 32);
    e[i] = lane ? pv : 0.f;
  }
  // walk own chunks writing per-chunk entry states
  for (int j = 0; j < 8; ++j) {
    float* ep = ws + E_OFF + (long)(c0 + j) * 6;
    const float* hp = ws + H_OFF + (long)(c0 + j) * 6;
    float nb[6];
    for (int i = 0; i < 6; ++i) ep[i] = e[i];
    for (int i = 0; i < 6; ++i) {
      float acc = hp[i];
      for (int k = 0; k < 6; ++k) acc += G[i * 6 + k] * e[k];
      nb[i] = acc;
    }
    for (int i = 0; i < 6; ++i) e[i] = nb[i];
  }
}

// k3: full blocked filter from true entry states; accumulate sum|y| per wave
__global__ __launch_bounds__(256) void lp_filter_abs(const float* __restrict__ xo,
                                                     const float* __restrict__ xt,
                                                     float* __restrict__ ws) {
  const int wid  = (blockIdx.x * blockDim.x + threadIdx.x) >> 5;
  const int lane = threadIdx.x & 31;
  const int col  = lane & 15;
  const int hi   = lane >> 4;
  const int row  = wid / WAVES_PER_ROW;
  const int w    = wid % WAVES_PER_ROW;

  v2f AY[6], AV[6];
  {
    const float* tu = ws + TU_OFF;
    const float* vw = ws + VW_OFF;
    #pragma unroll
    for (int wm = 0; wm < 6; ++wm) {
      int k0 = 4 * wm + 2 * hi;
      AY[wm].x = tu[col * 24 + k0];
      AY[wm].y = tu[col * 24 + k0 + 1];
      AV[wm].x = vw[col * 24 + k0];
      AV[wm].y = vw[col * 24 + k0 + 1];
    }
  }
  const long base = (long)row * TLEN + (long)(w * 16 + col) * LCHUNK;
  const float* po = xo + base;
  const float* pt = xt + base;

  float s0 = 0.f, s1 = 0.f, s2 = 0.f, s3 = 0.f, s4 = 0.f, s5 = 0.f;
  if (lane < 16) {
    long c = (long)row * NCHUNK_ROW + w * 16 + col;
    const float* ep = ws + E_OFF + c * 6;
    s0 = ep[0]; s1 = ep[1]; s2 = ep[2]; s3 = ep[3]; s4 = ep[4]; s5 = ep[5];
  }

  float acc = 0.f;
  for (int t = 0; t < TILES; ++t) {
    const int tb = t * 16;
    v2f B[6];
    #pragma unroll
    for (int wm = 0; wm < 4; ++wm) {
      int k0 = tb + 4 * wm + 2 * hi;
      B[wm].x = po[k0]     - pt[k0];
      B[wm].y = po[k0 + 1] - pt[k0 + 1];
    }
    float t0 = __shfl(s0, col, 32), t1 = __shfl(s1, col, 32), t2 = __shfl(s2, col, 32);
    float t3 = __shfl(s3, col, 32), t4 = __shfl(s4, col, 32), t5 = __shfl(s5, col, 32);
    B[4].x = hi ? t2 : t0;   B[4].y = hi ? t3 : t1;
    B[5].x = hi ? 0.f : t4;  B[5].y = hi ? 0.f : t5;

    v8f y = {0.f, 0.f, 0.f, 0.f, 0.f, 0.f, 0.f, 0.f};
    v8f d = {0.f, 0.f, 0.f, 0.f, 0.f, 0.f, 0.f, 0.f};
    #pragma unroll
    for (int wm = 0; wm < 6; ++wm) {
      y = __builtin_amdgcn_wmma_f32_16x16x4_f32(false, AY[wm], false, B[wm],
                                                (short)0, y, false, false);
      d = __builtin_amdgcn_wmma_f32_16x16x4_f32(false, AV[wm], false, B[wm],
                                                (short)0, d, false, false);
    }
    acc += fabsf(y[0]) + fabsf(y[1]) + fabsf(y[2]) + fabsf(y[3]) +
           fabsf(y[4]) + fabsf(y[5]) + fabsf(y[6]) + fabsf(y[7]);
    s0 = d[0]; s1 = d[1]; s2 = d[2]; s3 = d[3]; s4 = d[4]; s5 = d[5];
  }
  // wave32 butterfly reduction (deterministic)
  #pragma unroll
  for (int o = 16; o > 0; o >>= 1) acc += __shfl_xor(acc, o, 32);
  if (lane == 0) ws[P_OFF + wid] = acc;
}

// k4: deterministic tree reduce over per-wave partials -> mean
__global__ __launch_bounds__(256) void lp_reduce(const float* __restrict__ ws,
                                                 float* __restrict__ outp) {
  __shared__ float red[256];
  float a = 0.f;
  for (int i = threadIdx.x; i < NWAVES; i += 256) a += ws[P_OFF + i];
  red[threadIdx.x] = a;
  __syncthreads();
  for (int o = 128; o > 0; o >>= 1) {
    if ((int)threadIdx.x < o) red[threadIdx.x] += red[threadIdx.x + o];
    __syncthreads();
  }
  if (threadIdx.x == 0)
    outp[0] = red[0] * (1.0f / ((float)ROWS * (float)TLEN));
}

extern "C" void kernel_launch(void* const* d_in, const int* in_sizes, int n_in,
                              void* d_out, int out_size, void* d_ws, size_t ws_size,
                              hipStream_t stream) {
  (void)in_sizes; (void)n_in; (void)out_size; (void)ws_size;
  const float* xo = (const float*)d_in[0];  // output [96,131072] f32
  const float* xt = (const float*)d_in[1];  // target [96,131072] f32
  float* ws = (float*)d_ws;                 // needs ~1.19 MB
  float* op = (float*)d_out;                // scalar f32 loss

  lp_setup<<<1, 1, 0, stream>>>(ws);
  lp_chunkstate<<<NWAVES / 8, 256, 0, stream>>>(xo, xt, ws);   // 1536 waves
  lp_scan<<<ROWS, 32, 0, stream>>>(ws);                        // 96 waves
  lp_filter_abs<<<NWAVES / 8, 256, 0, stream>>>(xo, xt, ws);   // 1536 waves
  lp_reduce<<<1, 256, 0, stream>>>(ws, op);
}